// MixedActivation_25529285607531
// MI455X (gfx1250) — compile-verified
//
#include <hip/hip_runtime.h>

// MixedActivation over [rows, 48] f32: col%6 in {0,1,2} -> x*x, else PReLU
// with slope a[(col%6)-3]. Memory-bound: 384 MB traffic -> ~16.5 us floor at
// 23.3 TB/s HBM. Strategy: each thread owns 12 consecutive floats (two full
// 6-column activation periods) as 3 x 128-bit non-temporal loads/stores, so
// every lane's activation pattern is compile-time constant -- no mod, no
// compares, no cndmasks. VALU body is 6 muls + 6 (max,min,fma) per 96 bytes.

typedef __attribute__((ext_vector_type(4))) float v4f;

__device__ __forceinline__ float sq_act(float x) { return x * x; }

__device__ __forceinline__ float prelu_act(float x, float a) {
    // max(x,0) + a*min(x,0); a lives in an SGPR -> single v_fma operand
    return fmaxf(x, 0.0f) + a * fminf(x, 0.0f);
}

__global__ __launch_bounds__(256) void MixedActivation_25529285607531_kernel(
    const v4f* __restrict__ x4, const float* __restrict__ prelu_a,
    v4f* __restrict__ out4, int nthreads)
{
    const int t = blockIdx.x * blockDim.x + threadIdx.x;
    if (t >= nthreads) return;

    // Uniform scalars -> SGPRs via s_load
    const float a0 = prelu_a[0];
    const float a1 = prelu_a[1];
    const float a2 = prelu_a[2];

    const v4f* xin = x4 + (size_t)t * 3;
    v4f X0 = __builtin_nontemporal_load(xin + 0);
    v4f X1 = __builtin_nontemporal_load(xin + 1);
    v4f X2 = __builtin_nontemporal_load(xin + 2);

    // Columns (mod 6) covered by this thread: 0,1,2,3 | 4,5,0,1 | 2,3,4,5
    v4f R0, R1, R2;
    R0[0] = sq_act(X0[0]);
    R0[1] = sq_act(X0[1]);
    R0[2] = sq_act(X0[2]);
    R0[3] = prelu_act(X0[3], a0);

    R1[0] = prelu_act(X1[0], a1);
    R1[1] = prelu_act(X1[1], a2);
    R1[2] = sq_act(X1[2]);
    R1[3] = sq_act(X1[3]);

    R2[0] = sq_act(X2[0]);
    R2[1] = prelu_act(X2[1], a0);
    R2[2] = prelu_act(X2[2], a1);
    R2[3] = prelu_act(X2[3], a2);

    v4f* o = out4 + (size_t)t * 3;
    __builtin_nontemporal_store(R0, o + 0);
    __builtin_nontemporal_store(R1, o + 1);
    __builtin_nontemporal_store(R2, o + 2);
}

extern "C" void kernel_launch(void* const* d_in, const int* in_sizes, int n_in,
                              void* d_out, int out_size, void* d_ws, size_t ws_size,
                              hipStream_t stream) {
    (void)n_in; (void)d_ws; (void)ws_size; (void)out_size;

    const v4f*   x       = (const v4f*)d_in[0];
    const float* prelu_a = (const float*)d_in[1];
    v4f*         out     = (v4f*)d_out;

    const int n        = in_sizes[0];   // rows * 48 -> always divisible by 12
    const int nthreads = n / 12;        // 12 floats (3 x float4) per thread

    const int threads = 256;
    const int blocks  = (nthreads + threads - 1) / threads;

    MixedActivation_25529285607531_kernel<<<blocks, threads, 0, stream>>>(
        x, prelu_a, out, nthreads);
}